// TropConv2D_21947282883034
// MI455X (gfx1250) — compile-verified
//
#include <hip/hip_runtime.h>
#include <float.h>

// ---------------- problem constants (from reference) ----------------
#define C_IN   32                 // input channels
#define F_OUT  64                 // filters
#define H_IN   32
#define W_IN   32
#define H_OUT  30                 // H_IN - 3 + 1
#define W_OUT  30
#define ROWF   (W_IN * C_IN)      // floats per input image row = 1024
#define WSTR   292                // padded LDS row stride (dwords) for wT: 292 % 4 == 0,
                                  // 292 % 64 == 36 -> optimal 2-cycle b128 bank pattern
#define NWO    8                  // wo values handled per thread
#define NTHR   256                // 8 waves (wave32)

#ifndef __has_builtin
#define __has_builtin(x) 0
#endif

#if __has_builtin(__builtin_amdgcn_global_load_async_to_lds_b128) && \
    __has_builtin(__builtin_amdgcn_global_load_async_to_lds_b32)
#define USE_ASYNC_LDS 1
#endif

typedef __attribute__((ext_vector_type(4))) int v4i;
#define AS_GLOBAL __attribute__((address_space(1)))
#define AS_LDS    __attribute__((address_space(3)))

#ifdef USE_ASYNC_LDS
__device__ __forceinline__ void async_copy_b128(const float* gsrc, float* ldst) {
    __builtin_amdgcn_global_load_async_to_lds_b128(
        (AS_GLOBAL v4i*)gsrc, (AS_LDS v4i*)ldst, 0, 0);
}
__device__ __forceinline__ void async_copy_b32(const float* gsrc, float* ldst) {
    __builtin_amdgcn_global_load_async_to_lds_b32(
        (AS_GLOBAL int*)gsrc, (AS_LDS int*)ldst, 0, 0);
}
#endif

__global__ __launch_bounds__(NTHR)
void trop_conv2d_kernel(const float* __restrict__ x,
                        const float* __restrict__ w,
                        float* __restrict__ out)
{
    // Transposed, padded weights: wT[f][k], 64*292*4 = 74,752 B
    __shared__ float wT[F_OUT * WSTR];
    // Three contiguous input rows: 3*1024*4 = 12,288 B
    __shared__ float xs[3 * ROWF];

    const int tid = threadIdx.x;
    const int b   = blockIdx.x / H_OUT;
    const int ho  = blockIdx.x % H_OUT;

    // rows ho..ho+2 are contiguous: one 12 KB block
    const float* xsrc = x + (size_t)(b * H_IN + ho) * ROWF;

#ifdef USE_ASYNC_LDS
    // ---- CDNA5 async global->LDS DMA (tracked with ASYNCcnt) ----
    // x rows: 768 x 16B chunks, 3 per thread
    #pragma unroll
    for (int i = 0; i < 3; ++i) {
        const int e = tid + i * NTHR;
        async_copy_b128(xsrc + 4 * e, xs + 4 * e);
    }
    // weights, transposed on the fly: global w[k*64+f] -> LDS wT[f*WSTR+k]
    #pragma unroll 8
    for (int i = 0; i < 72; ++i) {
        const int e = tid + i * NTHR;       // 0..18431
        const int k = e >> 6;
        const int f = e & 63;
        async_copy_b32(w + e, wT + f * WSTR + k);
    }
#if __has_builtin(__builtin_amdgcn_s_wait_asynccnt)
    __builtin_amdgcn_s_wait_asynccnt(0);
#else
    asm volatile("s_wait_asynccnt 0" ::: "memory");
#endif
#else
    // ---- synchronous fallback ----
    #pragma unroll
    for (int i = 0; i < 3; ++i) {
        const int e = tid + i * NTHR;
        *(float4*)(xs + 4 * e) = *(const float4*)(xsrc + 4 * e);
    }
    #pragma unroll 8
    for (int i = 0; i < 72; ++i) {
        const int e = tid + i * NTHR;
        wT[(e & 63) * WSTR + (e >> 6)] = w[e];
    }
#endif
    __syncthreads();

    // ---- compute: lane -> filter (coalesced), 8 wo per thread ----
    const int f = tid & 63;
    const int g = tid >> 6;                 // 0..3

    float vmax[NWO], vmin[NWO];
    int   col[NWO];                         // clamped column base (floats)
    #pragma unroll
    for (int i = 0; i < NWO; ++i) {
        vmax[i] = -FLT_MAX;
        vmin[i] =  FLT_MAX;
        const int wo = g + 4 * i;           // covers 0..31; 30,31 clamped
        col[i] = (wo < W_OUT ? wo : (W_OUT - 1)) * C_IN;
    }

    const float* wrow = wT + f * WSTR;      // 16B-aligned (292 % 4 == 0)

    #pragma unroll
    for (int dy = 0; dy < 3; ++dy) {
        #pragma unroll
        for (int dx = 0; dx < 3; ++dx) {
            const float* xr = xs + dy * ROWF + dx * C_IN;      // + col + c
            const float* wr = wrow + (dy * 3 + dx) * C_IN;     // k-base
            #pragma unroll 4
            for (int c = 0; c < C_IN; c += 4) {
                const float4 wv = *(const float4*)(wr + c);    // 1 ds_load_b128 / 32 s-values
                #pragma unroll
                for (int i = 0; i < NWO; ++i) {
                    const float4 p = *(const float4*)(xr + col[i] + c); // broadcast read
                    const float s0 = p.x + wv.x;
                    const float s1 = p.y + wv.y;
                    const float s2 = p.z + wv.z;
                    const float s3 = p.w + wv.w;
                    // shaped for v_max3_f32 / v_min3_f32
                    vmax[i] = fmaxf(fmaxf(vmax[i], fmaxf(s0, s1)), fmaxf(s2, s3));
                    vmin[i] = fminf(fminf(vmin[i], fminf(s0, s1)), fminf(s2, s3));
                }
            }
        }
    }

    // ---- store: lanes f consecutive -> 256B contiguous per wo ----
    float* orow = out + (size_t)(b * H_OUT + ho) * W_OUT * F_OUT + f;
    #pragma unroll
    for (int i = 0; i < NWO; ++i) {
        const int wo = g + 4 * i;
        if (wo < W_OUT)
            orow[(size_t)wo * F_OUT] = vmax[i] - vmin[i];
    }
}

extern "C" void kernel_launch(void* const* d_in, const int* in_sizes, int n_in,
                              void* d_out, int out_size, void* d_ws, size_t ws_size,
                              hipStream_t stream)
{
    const float* x = (const float*)d_in[0];   // (B,32,32,32) f32
    const float* w = (const float*)d_in[1];   // (288,64)     f32
    float* out     = (float*)d_out;           // (B,30,30,64) f32

    const int B = in_sizes[0] / (H_IN * W_IN * C_IN);   // = 8
    dim3 grid(B * H_OUT);
    dim3 block(NTHR);
    hipLaunchKernelGGL(trop_conv2d_kernel, grid, block, 0, stream, x, w, out);
}